// GLIF_10548439679139
// MI455X (gfx1250) — compile-verified
//
#include <hip/hip_runtime.h>

// Problem constants (from reference)
#define T_STEPS 8192
#define NN      1024

// Decomposition: 32 persistent workgroups, each owns 32 output neurons.
#define NWG     32
#define COLS    32            // columns (neurons) per workgroup
#define TILES   (COLS / 16)   // 2 WMMA N-tiles per workgroup
#define KSPLIT  4             // K split across waves
#define BLOCK   256           // 8 wave32 waves
#define KG      (NN / 4)      // 256 k-groups of 4 (WMMA K=4)
#define GP      (KG / 2)      // 128 k-group *pairs* (one b128 B record each)
#define GPPW    (GP / KSPLIT) // 32 pairs per wave

// LDS layout (floats): weights (paired B-operand swizzle) | I_add vector | partials
#define WSH_FLOATS (GP * TILES * 128)         // 32768 floats = 128 KB
#define SMEM_FLOATS (WSH_FLOATS + NN + KSPLIT * COLS)
#define SMEM_BYTES  (SMEM_FLOATS * sizeof(float))  // ~132.5 KB (CDNA5: 320 KB LDS/WGP)

typedef __attribute__((ext_vector_type(2))) float v2f;
typedef __attribute__((ext_vector_type(4))) float v4f;
typedef __attribute__((ext_vector_type(8))) float v8f;
typedef __attribute__((__vector_size__(4 * sizeof(int)))) int vi4;  // matches builtin param

#define WMMA_F32(a, b, c) \
    __builtin_amdgcn_wmma_f32_16x16x4_f32(false, (a), false, (b), (short)0, (c), false, false)

__device__ __forceinline__ void grid_barrier(unsigned* bar, unsigned gen) {
    // Monotonic counter barrier: generation `gen` complete when counter >= gen*NWG.
    if (threadIdx.x == 0) {
        __threadfence();  // make this WG's global stores device-visible
        __hip_atomic_fetch_add(bar, 1u, __ATOMIC_ACQ_REL, __HIP_MEMORY_SCOPE_AGENT);
        while (__hip_atomic_load(bar, __ATOMIC_ACQUIRE, __HIP_MEMORY_SCOPE_AGENT) <
               gen * (unsigned)NWG) {
            __builtin_amdgcn_s_sleep(1);
        }
    }
    __syncthreads();
    __threadfence();  // acquire side: invalidate stale WGP-cache lines
}

__global__ void __launch_bounds__(BLOCK, 1)
glif_scan_kernel(const float* __restrict__ x_in, const float* __restrict__ w,
                 const float* __restrict__ E_L, const float* __restrict__ tau_m,
                 const float* __restrict__ G_, const float* __restrict__ R_I,
                 const float* __restrict__ f_v, const float* __restrict__ f_I,
                 const float* __restrict__ dth_s, const float* __restrict__ b_s,
                 const float* __restrict__ a_v, const float* __restrict__ b_v,
                 const float* __restrict__ th_inf, const float* __restrict__ dV,
                 const float* __restrict__ I_A,
                 float* __restrict__ out_v, float* __restrict__ out_s,
                 float* __restrict__ pub,   // [2][NN] double-buffered I_add exchange
                 unsigned* __restrict__ bar)
{
    extern __shared__ __align__(16) float lds[];
    float* w_sh = lds;                 // [TILES][GP][32 lanes][4] paired B records
    float* a_sh = w_sh + WSH_FLOATS;   // [NN] current I_add vector (linear)
    float* p_sh = a_sh + NN;           // [KSPLIT][COLS] partial sums

    const int tid  = threadIdx.x;
    const int wg   = blockIdx.x;
    const int lane = tid & 31;         // wave32
    const int wv   = tid >> 5;
    const int col0 = wg * COLS;

    // ---- One-time: weight slice, swizzled as *paired* WMMA B operands.
    // Record (tile,gp) = 128 floats, element lane*4+q:
    //   g = 2*gp + (q>>1), p = q&1, k = 4*g + (lane>=16 ? 2 : 0) + p
    //   value = w_masked[k][col0 + 16*tile + (lane&15)]
    // One ds_load_b128 per lane -> B operands (xy: even g, zw: odd g) for 2 WMMAs.
    for (int f = tid; f < WSH_FLOATS; f += BLOCK) {
        int tile = f >> 14;            // / (GP*128)
        int r    = f & 16383;
        int gp   = r >> 7;
        int e    = r & 127;
        int ln   = e >> 2;
        int q    = e & 3;
        int g    = 2 * gp + (q >> 1);
        int p    = q & 1;
        int k    = 4 * g + ((ln & 16) ? 2 : 0) + p;
        int c    = col0 + tile * 16 + (ln & 15);
        float val = w[(size_t)k * NN + c];
        if (k == c) val = 0.0f;        // self-recurrence mask (1 - eye)
        w_sh[f] = val;
    }

    // ---- Per-neuron parameters + state in registers (threads 0..COLS-1)
    const int mycol = col0 + tid;
    float r_EL = 0, r_itau = 1, r_G = 0, r_RI = 0, r_fv = 0, r_fI = 0, r_dts = 0,
          r_bs = 0, r_av = 0, r_bv = 0, r_ti = 0, r_dV = 0, r_IA = 0;
    float v = 0, ths = 0, thv = 0, Iadd = 0;
    if (tid < COLS) {
        r_EL = E_L[mycol];    r_itau = 1.0f / tau_m[mycol]; r_G  = G_[mycol];
        r_RI = R_I[mycol];    r_fv   = f_v[mycol];          r_fI = f_I[mycol];
        r_dts = dth_s[mycol]; r_bs   = b_s[mycol];          r_av = a_v[mycol];
        r_bv = b_v[mycol];    r_ti   = th_inf[mycol];       r_dV = dV[mycol];
        r_IA = I_A[mycol];
        v = r_EL; ths = 30.0f; thv = 1.0f; Iadd = 0.0f;
        pub[mycol] = 0.0f;             // I_add(t=0) = 0 in buffer 0
    }
    __syncthreads();
    grid_barrier(bar, 1u);             // buffer 0 fully initialized device-wide

    const int tile = wv & (TILES - 1);
    const int kq   = wv >> 1;                  // 0..KSPLIT-1
    const int gp0  = kq * GPPW;
    const int ap   = (lane >> 4) << 1;         // lanes<16 -> +0, lanes>=16 -> +2
    const float* bbase = w_sh + (size_t)tile * (GP * 128) + lane * 4;

    for (int t = 0; t < T_STEPS; ++t) {
        // ---- Gather full I_add vector (prev step) from L2 into LDS (4 KB).
#if __has_builtin(__builtin_amdgcn_global_load_async_to_lds_b128)
        // CDNA5 async DMA path: no VGPR round-trip, tracked by ASYNCcnt.
        __builtin_amdgcn_global_load_async_to_lds_b128(
            (__attribute__((address_space(1))) vi4*)(pub + (size_t)(t & 1) * NN + tid * 4),
            (__attribute__((address_space(3))) vi4*)(a_sh + tid * 4),
            0, 0);
    #if __has_builtin(__builtin_amdgcn_s_wait_asynccnt)
        __builtin_amdgcn_s_wait_asynccnt(0);
    #else
        asm volatile("s_wait_asynccnt 0x0" ::: "memory");
    #endif
#else
        ((v4f*)a_sh)[tid] = ((const v4f*)(pub + (size_t)(t & 1) * NN))[tid];
#endif
        float xval = 0.0f;
        if (tid < COLS) {
            xval = x_in[(size_t)t * NN + mycol];
            if (t + 1 < T_STEPS)
                __builtin_prefetch(&x_in[(size_t)(t + 1) * NN + mycol], 0, 0);
        }
        __syncthreads();

        // ---- Matvec slice via V_WMMA_F32_16X16X4_F32, 2-deep software pipeline.
        // Even-g WMMA k-map: {8gp+0, 8gp+1, 8gp+2, 8gp+3}
        // Odd -g WMMA k-map: {8gp+4, 8gp+5, 8gp+6, 8gp+7}
        v8f acc0 = {};
        v8f acc1 = {};
        v2f a0_0 = *(const v2f*)&a_sh[8 * gp0 + ap];
        v2f a1_0 = *(const v2f*)&a_sh[8 * gp0 + 4 + ap];
        v4f b_0  = *(const v4f*)(bbase + (size_t)gp0 * 128);
        v2f a0_1 = *(const v2f*)&a_sh[8 * (gp0 + 1) + ap];
        v2f a1_1 = *(const v2f*)&a_sh[8 * (gp0 + 1) + 4 + ap];
        v4f b_1  = *(const v4f*)(bbase + (size_t)(gp0 + 1) * 128);
        #pragma unroll 6
        for (int i = 0; i < GPPW - 2; ++i) {
            const int gpn = gp0 + i + 2;
            v2f a0_2 = *(const v2f*)&a_sh[8 * gpn + ap];
            v2f a1_2 = *(const v2f*)&a_sh[8 * gpn + 4 + ap];
            v4f b_2  = *(const v4f*)(bbase + (size_t)gpn * 128);
            acc0 = WMMA_F32(a0_0, b_0.xy, acc0);
            acc1 = WMMA_F32(a1_0, b_0.zw, acc1);
            a0_0 = a0_1; a1_0 = a1_1; b_0 = b_1;
            a0_1 = a0_2; a1_1 = a1_2; b_1 = b_2;
        }
        acc0 = WMMA_F32(a0_0, b_0.xy, acc0);
        acc1 = WMMA_F32(a1_0, b_0.zw, acc1);
        acc0 = WMMA_F32(a0_1, b_1.xy, acc0);
        acc1 = WMMA_F32(a1_1, b_1.zw, acc1);

        // All D rows identical (A rows replicated): element 0 = result[col].
        if (lane < 16) p_sh[kq * COLS + tile * 16 + lane] = acc0[0] + acc1[0];
        __syncthreads();

        // ---- Elementwise GLIF update for this WG's neurons.
        if (tid < COLS) {
            float I = p_sh[tid] + p_sh[COLS + tid] + p_sh[2 * COLS + tid] +
                      p_sh[3 * COLS + tid] + 0.85f * xval;
            float v_next = v + (I * r_RI - r_G * (v - r_EL)) * r_itau;
            float thr    = ths + thv;
            float ssoft  = 1.0f / (1.0f + __expf(-(v_next - thr)));
            float spiked = (v_next >= thr) ? 1.0f : 0.0f;
            float nsp    = 1.0f - spiked;
            float v_rst  = r_EL + r_fv * (v - r_EL) - r_dV;
            float v_new  = spiked * v_rst + nsp * v_next;
            ths = (1.0f - r_bs) * ths + spiked * r_dts;
            float dthv = r_av * (v_new - r_EL) - r_bv * (thv - r_ti);
            thv = thv + nsp * dthv;
            Iadd = (1.0f - r_fI) * Iadd + ssoft * r_IA;
            v = v_new;

            out_v[(size_t)t * NN + mycol] = v_new;
            out_s[(size_t)t * NN + mycol] = ssoft;
            pub[(size_t)((t + 1) & 1) * NN + mycol] = Iadd;  // publish for next step
        }
        __syncthreads();
        grid_barrier(bar, (unsigned)(t + 2));
    }
}

extern "C" void kernel_launch(void* const* d_in, const int* in_sizes, int n_in,
                              void* d_out, int out_size, void* d_ws, size_t ws_size,
                              hipStream_t stream) {
    const float* x_in  = (const float*)d_in[0];
    const float* w     = (const float*)d_in[1];
    const float* E_L   = (const float*)d_in[2];
    const float* tau_m = (const float*)d_in[3];
    const float* G_    = (const float*)d_in[4];
    const float* R_I   = (const float*)d_in[5];
    const float* f_v   = (const float*)d_in[6];
    const float* f_I   = (const float*)d_in[7];
    const float* dth_s = (const float*)d_in[8];
    const float* b_s   = (const float*)d_in[9];
    const float* a_v   = (const float*)d_in[10];
    const float* b_v   = (const float*)d_in[11];
    const float* th_in = (const float*)d_in[12];
    const float* dV    = (const float*)d_in[13];
    const float* I_A   = (const float*)d_in[14];

    float* out_v = (float*)d_out;                       // vs     [T, N]
    float* out_s = out_v + (size_t)T_STEPS * NN;        // spikes [T, N]

    unsigned* bar = (unsigned*)d_ws;                    // barrier counter
    float* pub = (float*)((char*)d_ws + 256);           // [2][NN] I_add exchange

    // Barrier counter must start at 0 every call (capture-safe async memset).
    (void)hipMemsetAsync(bar, 0, sizeof(unsigned), stream);

    // ~132.5 KB dynamic LDS per WG (CDNA5: up to 320 KB/WGP).
    (void)hipFuncSetAttribute((const void*)glif_scan_kernel,
                              hipFuncAttributeMaxDynamicSharedMemorySize,
                              (int)SMEM_BYTES);

    glif_scan_kernel<<<NWG, BLOCK, SMEM_BYTES, stream>>>(
        x_in, w, E_L, tau_m, G_, R_I, f_v, f_I, dth_s, b_s, a_v, b_v, th_in,
        dV, I_A, out_v, out_s, pub, bar);
}